// RotationVQ_25589415150076
// MI455X (gfx1250) — compile-verified
//
#include <hip/hip_runtime.h>

typedef float v2f __attribute__((ext_vector_type(2)));
typedef float v8f __attribute__((ext_vector_type(8)));
typedef int   v4i __attribute__((vector_size(16)));   // matches async-LDS builtin param type

#define DIMK 64
#define NCODES 8192
#define NROWS 16384
#define ROWS_PER_WG 32
#define CHUNK 64
#define NCHUNK (NCODES / CHUNK)
#define LSTR 68   // padded LDS row stride (floats): bank = (4*row + d) % 64 -> conflict-free b64 frag reads

#define ZQ_ELEMS (16 * 64 * 32 * 32)   // 1048576
#define IDX_OFF  ZQ_ELEMS
#define COMMIT_OFF (ZQ_ELEMS + NROWS)

#if defined(__has_builtin)
#if __has_builtin(__builtin_amdgcn_global_load_async_to_lds_b128) && \
    __has_builtin(__builtin_amdgcn_s_wait_asynccnt)
#define USE_ASYNC 1
#endif
#endif
#ifndef USE_ASYNC
#define USE_ASYNC 0
#endif

// Move one 64x64 f32 codebook chunk (16 KB) global -> LDS with 256 threads,
// 4 x b128 per thread. Async path: no VGPR data, tracked by ASYNCcnt.
__device__ __forceinline__ void stage_chunk(const float* __restrict__ emb,
                                            float* __restrict__ dst, int t, int cidx) {
  const float* g = emb + (size_t)cidx * (CHUNK * DIMK);
#pragma unroll
  for (int j = 0; j < 4; ++j) {
    int li = (j * 256 + t) * 4;
    float* l = &dst[(li >> 6) * LSTR + (li & 63)];
#if USE_ASYNC
    __builtin_amdgcn_global_load_async_to_lds_b128(
        (__attribute__((address_space(1))) v4i*)(g + li),
        (__attribute__((address_space(3))) v4i*)l, 0, 0);
#else
    *(float4*)l = *(const float4*)(g + li);
#endif
  }
}

// ---------------------------------------------------------------- b2[k] = ||e_k||^2
__global__ __launch_bounds__(256) void vq_b2_kernel(const float* __restrict__ emb,
                                                    float* __restrict__ b2) {
  int k = blockIdx.x * 256 + threadIdx.x;
  if (k < NCODES) {
    const float4* p = (const float4*)(emb + (size_t)k * DIMK);
    float s = 0.f;
#pragma unroll
    for (int j = 0; j < 16; ++j) {
      float4 e = p[j];
      s += e.x * e.x + e.y * e.y + e.z * e.z + e.w * e.w;
    }
    b2[k] = s;
  }
}

// ---------------------------------------------------------------- main WMMA argmin GEMM
__global__ __launch_bounds__(256) void vq_argmin_kernel(const float* __restrict__ z,
                                                        const float* __restrict__ emb,
                                                        const float* __restrict__ b2,
                                                        int* __restrict__ out_idx) {
  __shared__ float zl[ROWS_PER_WG * LSTR];   // 32 rows x 64 depth (padded)
  __shared__ float el[2 * CHUNK * LSTR];     // double-buffered 64 codes x 64 depth
  __shared__ float red_v[ROWS_PER_WG * 4];
  __shared__ int   red_i[ROWS_PER_WG * 4];

  const int t = threadIdx.x;
  const int wave = t >> 5;
  const int lane = t & 31;
  const int half = lane >> 4;   // lane group (selects K pair / row offset +8)
  const int l16  = lane & 15;
  const int rowTile = wave >> 2;   // 0..1
  const int colSub  = wave & 3;    // 0..3

  const int wgRow0 = blockIdx.x * ROWS_PER_WG;
  const int bimg = wgRow0 >> 10;     // 1024 rows per image, 32 | 1024
  const int hw0  = wgRow0 & 1023;

  // Kick off async staging of codebook chunk 0 immediately.
  stage_chunk(emb, el, t, 0);

  // Stage z tile: z_flat[n,d] = z[b, d, hw] (strided in global, row-major in LDS)
#pragma unroll
  for (int i = 0; i < (ROWS_PER_WG * DIMK) / 256; ++i) {
    int li = i * 256 + t;
    int d  = li >> 5;
    int nl = li & 31;
    zl[nl * LSTR + d] = z[(size_t)bimg * (DIMK * 1024) + (size_t)d * 1024 + hw0 + nl];
  }
  __syncthreads();

  // Persistent A fragments, pre-scaled by -2 so acc = b2 - 2*z.e directly.
  // (ISA 7.12.2: A 16x4 fp32 -> lane%16 = M, lane/16 selects K pair, .x/.y = K lo/hi)
  v2f a[16];
  const int m = rowTile * 16 + l16;
#pragma unroll
  for (int s = 0; s < 16; ++s) {
    int d = 4 * s + 2 * half;
    v2f av = *(const v2f*)&zl[m * LSTR + d];
    a[s].x = -2.0f * av.x;
    a[s].y = -2.0f * av.y;
  }

  float minv[8];
  int   mini[8];
#pragma unroll
  for (int v = 0; v < 8; ++v) { minv[v] = 3.4e38f; mini[v] = 0; }

  const int cloc = colSub * 16 + l16;
  for (int i = 0; i < NCHUNK; ++i) {
    float* eb = el + (i & 1) * (CHUNK * LSTR);
    const int code = i * CHUNK + colSub * 16 + l16;  // this lane's output column
    const float bb = b2[code];                       // issue early, hidden by wait/barrier

#if USE_ASYNC
    __builtin_amdgcn_s_wait_asynccnt(0);             // my chunk-i DMA done
#endif
    __syncthreads();                                 // everyone's chunk-i done;
                                                     // all waves finished chunk i-1 compute
    if (i + 1 < NCHUNK) {
      // Post-barrier: buffer (i+1)&1 is dead; DMA overlaps the WMMA chain below.
      stage_chunk(emb, el + ((i + 1) & 1) * (CHUNK * LSTR), t, i + 1);
    }

    // acc starts at ||e||^2 (uniform over the lane's 8 rows); A carries the -2 factor
    v8f acc = {bb, bb, bb, bb, bb, bb, bb, bb};
#pragma unroll
    for (int s = 0; s < 16; ++s) {
      int d = 4 * s + 2 * half;
      v2f bv = *(const v2f*)&eb[cloc * LSTR + d];
      acc = __builtin_amdgcn_wmma_f32_16x16x4_f32(
          false, a[s], false, bv, (short)0, acc, false, false);
    }

#pragma unroll
    for (int v = 0; v < 8; ++v) {                    // row M = v + 8*half
      if (acc[v] < minv[v]) { minv[v] = acc[v]; mini[v] = code; }  // ascending -> keeps first
    }
  }

  // Cross-lane argmin within each 16-lane half (same rows, 16 columns)
#pragma unroll
  for (int v = 0; v < 8; ++v) {
    float val = minv[v];
    int   idx = mini[v];
#pragma unroll
    for (int off = 1; off < 16; off <<= 1) {
      float ov = __shfl_xor(val, off, 16);
      int   oi = __shfl_xor(idx, off, 16);
      if (ov < val || (ov == val && oi < idx)) { val = ov; idx = oi; }
    }
    if (l16 == 0) {
      int rloc = rowTile * 16 + v + 8 * half;  // 0..31 local row
      red_v[rloc * 4 + colSub] = val;
      red_i[rloc * 4 + colSub] = idx;
    }
  }
  __syncthreads();

  if (t < ROWS_PER_WG) {
    float val = red_v[t * 4];
    int   idx = red_i[t * 4];
#pragma unroll
    for (int c = 1; c < 4; ++c) {
      float ov = red_v[t * 4 + c];
      int   oi = red_i[t * 4 + c];
      if (ov < val || (ov == val && oi < idx)) { val = ov; idx = oi; }
    }
    out_idx[wgRow0 + t] = idx;
  }
}

// ---------------------------------------------------------------- gather q, scatter z_q, partial MSE
__global__ __launch_bounds__(256) void vq_finalize_kernel(const float* __restrict__ z,
                                                          const float* __restrict__ emb,
                                                          const int* __restrict__ idxs,
                                                          float* __restrict__ out,
                                                          float* __restrict__ partial) {
  __shared__ float sred[256];
  const int t  = threadIdx.x;
  const int nl = t & 63;
  const int n  = blockIdx.x * 64 + nl;
  const int bimg = n >> 10;
  const int hw   = n & 1023;
  const int idx  = idxs[n];

  float s = 0.f;
#pragma unroll
  for (int i = 0; i < 16; ++i) {
    int d = (t >> 6) + 4 * i;
    float q = emb[(size_t)idx * DIMK + d];
    size_t off = (size_t)bimg * (DIMK * 1024) + (size_t)d * 1024 + hw;
    float zv = z[off];
    out[off] = q;                        // forward value of z_q is exactly q
    float df = zv - q;
    s += df * df;
  }
  if (t < 64) out[IDX_OFF + n] = (float)idx;

  sred[t] = s;
  __syncthreads();
  for (int st = 128; st > 0; st >>= 1) {
    if (t < st) sred[t] += sred[t + st];
    __syncthreads();
  }
  if (t == 0) partial[blockIdx.x] = sred[0];
}

__global__ __launch_bounds__(256) void vq_commit_kernel(const float* __restrict__ partial,
                                                        float* __restrict__ out) {
  __shared__ float sred[256];
  const int t = threadIdx.x;
  sred[t] = partial[t];
  __syncthreads();
  for (int st = 128; st > 0; st >>= 1) {
    if (t < st) sred[t] += sred[t + st];
    __syncthreads();
  }
  if (t == 0) out[COMMIT_OFF] = sred[0] / (float)((size_t)NROWS * DIMK);
}

// ----------------------------------------------------------------
extern "C" void kernel_launch(void* const* d_in, const int* in_sizes, int n_in,
                              void* d_out, int out_size, void* d_ws, size_t ws_size,
                              hipStream_t stream) {
  const float* z   = (const float*)d_in[0];
  const float* emb = (const float*)d_in[1];
  float* out = (float*)d_out;

  char* ws = (char*)d_ws;
  int*   idxs    = (int*)ws;                          // 16384 * 4  = 64 KB
  float* b2      = (float*)(ws + 65536);              //  8192 * 4  = 32 KB
  float* partial = (float*)(ws + 65536 + 32768);      //   256 * 4

  vq_b2_kernel<<<NCODES / 256, 256, 0, stream>>>(emb, b2);
  vq_argmin_kernel<<<NROWS / ROWS_PER_WG, 256, 0, stream>>>(z, emb, b2, idxs);
  vq_finalize_kernel<<<NROWS / 64, 256, 0, stream>>>(z, emb, idxs, out, partial);
  vq_commit_kernel<<<1, 256, 0, stream>>>(partial, out);
}